// TokenRouter_46712064311616
// MI455X (gfx1250) — compile-verified
//
#include <hip/hip_runtime.h>

typedef __attribute__((ext_vector_type(2))) float v2f;
typedef __attribute__((ext_vector_type(8))) float v8f;

namespace {
constexpr int kBT  = 16384;   // B*T tokens
constexpr int kD   = 4096;    // hidden dim
constexpr int kE   = 64;      // experts
constexpr int kK   = 2;       // top-k
constexpr int kKC  = 64;      // K-chunk staged in LDS (double-buffered)
constexpr int kSteps = kKC / 4;       // 16 wmma k-steps per chunk (K=4 each)
constexpr int kQ4   = kKC / 4;        // float4s per expert row per chunk
constexpr int kWStride = kKC + 4;     // 68 floats: pad -> conflict-free ds reads
constexpr int kChunks = kD / kKC;     // 64
constexpr int kTokPerBlk = 64;        // 4 waves x 16-token M tiles
constexpr int kThreads = 128;
constexpr int kLStride = 68;          // logits row stride (16B aligned, padded)
}

// Async global -> LDS copy (CDNA5, ASYNCcnt path). LDS operand is the raw
// group-segment byte offset = low 32 bits of the generic pointer.
__device__ __forceinline__ void async_copy_b128(const float* gsrc, float* ldst) {
    unsigned lofs = (unsigned)(uintptr_t)ldst;
    unsigned long long ga = (unsigned long long)(uintptr_t)gsrc;
    asm volatile("global_load_async_to_lds_b128 %0, %1, off"
                 :: "v"(lofs), "v"(ga)
                 : "memory");
}

__device__ __forceinline__ void wait_async0() {
    asm volatile("s_wait_asynccnt 0x0" ::: "memory");
}

__global__ __launch_bounds__(kThreads)
void token_router_kernel(const float* __restrict__ x,
                         const float* __restrict__ W,
                         float* __restrict__ out_probs,
                         int* __restrict__ out_idx,
                         float* __restrict__ out_wts)
{
    __shared__ float lds_w[2][kE * kWStride];       // 2 x ~17 KB W chunk buffers
    __shared__ float lds_l[kTokPerBlk * kLStride];  // ~17 KB logits

    const int tid  = threadIdx.x;
    const int wave = tid >> 5;
    const int lane = tid & 31;
    const int half = lane >> 4;   // 0: k,k+1   1: k+2,k+3 (16x4 f32 A layout)
    const int lr   = lane & 15;

    const int tok_base = blockIdx.x * kTokPerBlk;
    const int my_tok   = tok_base + wave * 16 + lr;

    v8f acc0 = {}; v8f acc1 = {}; v8f acc2 = {}; v8f acc3 = {};

    const float* rowp = x + (size_t)my_tok * kD;   // this lane's token row
    const float* aptr = rowp + 2 * half;

    // ---- prologue: async-stage chunk 0 into buffer 0 ----
    #pragma unroll
    for (int p = 0; p < (kE * kQ4) / kThreads; ++p) {   // 8 asyncs / thread
        int idx = p * kThreads + tid;
        int n   = idx >> 4;        // expert row 0..63
        int kq  = idx & 15;        // float4 within chunk
        async_copy_b128(W + (size_t)n * kD + 4 * kq,
                        &lds_w[0][n * kWStride + 4 * kq]);
    }
    wait_async0();
    __syncthreads();

    for (int c = 0; c < kChunks; ++c) {
        const int k0 = c * kKC;
        const float* cur = lds_w[c & 1];

        if (c + 1 < kChunks) {
            // async-stage next chunk into the other buffer (its last readers
            // finished at the barrier that ended iteration c-1)
            float* nxt = lds_w[(c + 1) & 1];
            const int kn = k0 + kKC;
            #pragma unroll
            for (int p = 0; p < (kE * kQ4) / kThreads; ++p) {
                int idx = p * kThreads + tid;
                int n   = idx >> 4;
                int kq  = idx & 15;
                async_copy_b128(W + (size_t)n * kD + kn + 4 * kq,
                                &nxt[n * kWStride + 4 * kq]);
            }
            // prefetch next chunk of this lane's x row (2 lines per row,
            // split across the two half-waves) -> global_prefetch_b8
            __builtin_prefetch(rowp + kn + half * 32, 0, 3);
        }

        #pragma unroll 4
        for (int s = 0; s < kSteps; ++s) {
            v2f a = *(const v2f*)(aptr + k0 + 4 * s);
            const int bofs = 4 * s + 2 * half;
            v2f b0 = *(const v2f*)&cur[( 0 + lr) * kWStride + bofs];
            v2f b1 = *(const v2f*)&cur[(16 + lr) * kWStride + bofs];
            v2f b2 = *(const v2f*)&cur[(32 + lr) * kWStride + bofs];
            v2f b3 = *(const v2f*)&cur[(48 + lr) * kWStride + bofs];
            acc0 = __builtin_amdgcn_wmma_f32_16x16x4_f32(false, a, false, b0, (short)0, acc0, false, false);
            acc1 = __builtin_amdgcn_wmma_f32_16x16x4_f32(false, a, false, b1, (short)0, acc1, false, false);
            acc2 = __builtin_amdgcn_wmma_f32_16x16x4_f32(false, a, false, b2, (short)0, acc2, false, false);
            acc3 = __builtin_amdgcn_wmma_f32_16x16x4_f32(false, a, false, b3, (short)0, acc3, false, false);
        }

        wait_async0();     // my writes into the next buffer are in LDS
        __syncthreads();   // everyone's writes visible; reads of cur retired
    }

    // Scatter C fragments (16x16 f32 layout: VGPR r -> M = r + 8*half, N = lr)
    // to token-major LDS rows.
    #pragma unroll
    for (int r = 0; r < 8; ++r) {
        int tl = wave * 16 + half * 8 + r;
        float* row = &lds_l[tl * kLStride];
        row[ 0 + lr] = acc0[r];
        row[16 + lr] = acc1[r];
        row[32 + lr] = acc2[r];
        row[48 + lr] = acc3[r];
    }
    __syncthreads();

    // One thread per token: softmax + top-2 (+ renormalized weights).
    if (tid < kTokPerBlk) {
        float* l = &lds_l[tid * kLStride];
        float m1 = -__builtin_inff(); int i1 = 0;
        float m2 = -__builtin_inff(); int i2 = 0;
        #pragma unroll
        for (int j = 0; j < kE; ++j) {
            float v = l[j];
            if (v > m1)      { m2 = m1; i2 = i1; m1 = v; i1 = j; }
            else if (v > m2) { m2 = v;  i2 = j; }
        }
        float sum = 0.f;
        #pragma unroll
        for (int j = 0; j < kE; ++j) {
            float e = __expf(l[j] - m1);
            l[j] = e;
            sum += e;
        }
        const float inv = 1.0f / sum;
        const int gtok = tok_base + tid;
        float4* po = (float4*)(out_probs + (size_t)gtok * kE);
        #pragma unroll
        for (int q = 0; q < kE / 4; ++q) {
            float4 ev = *(float4*)&l[4 * q];
            po[q] = make_float4(ev.x * inv, ev.y * inv, ev.z * inv, ev.w * inv);
        }
        const float p1 = inv;                    // exp(m1-m1)/sum
        const float p2 = __expf(m2 - m1) * inv;
        const float dn = p1 + p2 + 1e-9f;
        out_idx[gtok * kK + 0] = i1;
        out_idx[gtok * kK + 1] = i2;
        out_wts[gtok * kK + 0] = p1 / dn;
        out_wts[gtok * kK + 1] = p2 / dn;
    }
}

extern "C" void kernel_launch(void* const* d_in, const int* in_sizes, int n_in,
                              void* d_out, int out_size, void* d_ws, size_t ws_size,
                              hipStream_t stream) {
    (void)in_sizes; (void)n_in; (void)out_size; (void)d_ws; (void)ws_size;
    const float* x = (const float*)d_in[0];
    const float* W = (const float*)d_in[1];

    float* probs = (float*)d_out;                               // BT*E f32
    int*   idx   = (int*)(probs + (size_t)kBT * kE);            // BT*K i32 (bit-region)
    float* wts   = probs + (size_t)kBT * kE + (size_t)kBT * kK; // BT*K f32

    dim3 grid(kBT / kTokPerBlk);   // 256 blocks
    dim3 block(kThreads);          // 128 threads = 4 waves
    hipLaunchKernelGGL(token_router_kernel, grid, block, 0, stream,
                       x, W, probs, idx, wts);
}